// PerVertQuaternion_21397527069153
// MI455X (gfx1250) — compile-verified
//
#include <hip/hip_runtime.h>
#include <hip/hip_bf16.h>

#define BLOCK 256
#define CHUNK (BLOCK * 3) // dwords of face indices per chunk (3 KB)
#define NIT 4             // chunks per block (software-pipelined)

// ---- CDNA5 TDM availability -------------------------------------------------
#if defined(__gfx1250__) && __has_builtin(__builtin_amdgcn_tensor_load_to_lds) && \
    __has_builtin(__builtin_amdgcn_s_wait_tensorcnt)
#define USE_TDM 1
#else
#define USE_TDM 0
#endif

#if __has_include(<hip/amd_detail/amd_gfx1250_TDM.h>)
#define TDM_SIX_ARG 1 // therock-10.0 headers -> 6-arg builtin
#else
#define TDM_SIX_ARG 0 // ROCm 7.2 clang-22 -> 5-arg builtin
#endif

typedef __attribute__((ext_vector_type(4))) unsigned int u32x4;
typedef __attribute__((ext_vector_type(4))) int i32x4;
typedef __attribute__((ext_vector_type(8))) int i32x8;

// ---- small float3 helpers ---------------------------------------------------
struct F3 { float x, y, z; };
__device__ __forceinline__ F3 f3sub(F3 a, F3 b) { return {a.x - b.x, a.y - b.y, a.z - b.z}; }
__device__ __forceinline__ F3 f3cross(F3 a, F3 b) {
    return {a.y * b.z - a.z * b.y, a.z * b.x - a.x * b.z, a.x * b.y - a.y * b.x};
}
__device__ __forceinline__ float f3dot(F3 a, F3 b) { return a.x * b.x + a.y * b.y + a.z * b.z; }
__device__ __forceinline__ F3 f3nrm(F3 v) {
    float n = sqrtf(f3dot(v, v));
    float r = 1.0f / fmaxf(n, 1e-12f);
    return {v.x * r, v.y * r, v.z * r};
}
__device__ __forceinline__ F3 ldv(const float* __restrict__ p, int i) {
    const float* q = p + 3ull * (unsigned)i;
    return {q[0], q[1], q[2]};
}
__device__ __forceinline__ void triFrame(F3 a, F3 b, F3 c, F3& X, F3& Y, F3& Z) {
    F3 n = f3nrm(f3cross(f3sub(b, a), f3sub(c, a)));
    F3 d = f3sub(b, a);
    X = f3nrm(f3cross(d, n));
    Y = f3nrm(f3cross(d, X));
    Z = f3nrm(d);
}

#if USE_TDM
// Issue one TDM DMA: CHUNK dwords of the face-index stream -> LDS buffer.
// All descriptor words are wave-uniform (blockIdx-derived) -> SGPR descriptor.
// tensor_dim0 clamps to remaining dwords: TDM OOB reads return zero.
__device__ __forceinline__ void tdm_issue(const int* __restrict__ faces, long long dwBase,
                                          long long totalDw, unsigned ldsOff) {
    unsigned long long ga =
        (unsigned long long)(const void*)faces + (unsigned long long)dwBase * 4ull;
    long long rem = totalDw - dwBase;
    unsigned remDw = (rem >= (long long)CHUNK) ? (unsigned)CHUNK : (unsigned)rem;

    // D# group 0: count=1 | lds_addr | global_addr[56:0] | type=2
    u32x4 g0 = {1u, ldsOff, (unsigned)(ga & 0xFFFFFFFFull),
                (unsigned)((ga >> 32) & 0x01FFFFFFull) | 0x80000000u};
    // D# group 1: data_size=4B; tensor_dim0=remDw; tensor_dim1=1;
    //             tile_dim0=CHUNK; tile_dim1=1; dim0_stride=CHUNK
    i32x8 g1 = {(int)0x00020000,
                (int)((remDw & 0xFFFFu) << 16),
                (int)(((remDw >> 16) & 0xFFFFu) | (1u << 16)),
                (int)((unsigned)CHUNK << 16),
                1,
                (int)CHUNK,
                0, 0};
    i32x4 gz = {0, 0, 0, 0};
#if TDM_SIX_ARG
    i32x8 gz8 = {0, 0, 0, 0, 0, 0, 0, 0};
    __builtin_amdgcn_tensor_load_to_lds(g0, g1, gz, gz, gz8, 0);
#else
    __builtin_amdgcn_tensor_load_to_lds(g0, g1, gz, gz, 0);
#endif
}
#endif

// ---- kernel 1: zero the per-vertex accumulator ------------------------------
__global__ void pvq_zero_kernel(float* __restrict__ out, int n) {
    int i = blockIdx.x * blockDim.x + threadIdx.x;
    if (i < n) out[i] = 0.0f;
}

// ---- kernel 2: per-face quaternion + area-weighted scatter ------------------
// Each block processes NIT chunks of BLOCK faces with a double-buffered,
// TDM-software-pipelined index stage: DMA of chunk it+1 overlaps compute of
// chunk it.
__global__ void __launch_bounds__(BLOCK) pvq_face_kernel(
    const float* __restrict__ mesh, const float* __restrict__ cano,
    const int* __restrict__ faces, float* __restrict__ out, int F) {
    __shared__ int sIdx[2][CHUNK];
    const int tid = threadIdx.x;
    const long long totalDw = 3LL * F;
    const long long totalChunks = ((long long)F + BLOCK - 1) / BLOCK;
    const long long chunk0 = (long long)blockIdx.x * NIT;
    const int nit = (int)((totalChunks - chunk0 < NIT) ? (totalChunks - chunk0) : NIT);
    if (nit <= 0) return;

    // Escape the LDS array (input-only asm + memory clobber): the TDM writes it
    // behind the compiler's back, but reads stay in addrspace(3) -> ds_load.
    asm volatile("" ::"s"(&sIdx[0][0]) : "memory");

#if USE_TDM
    const unsigned ldsOff0 = (unsigned)(unsigned long long)(void*)&sIdx[0][0];
    const unsigned ldsOff1 = (unsigned)(unsigned long long)(void*)&sIdx[1][0];
    if (tid < 32) { // wave 0 drives the DMA pipeline (TENSORcnt is per-wave)
        tdm_issue(faces, chunk0 * CHUNK, totalDw, ldsOff0);
        if (nit > 1) tdm_issue(faces, (chunk0 + 1) * CHUNK, totalDw, ldsOff1);
    }
#endif

    for (int it = 0; it < nit; ++it) {
#if USE_TDM
        if (tid < 32) {
            if (it + 1 < nit) __builtin_amdgcn_s_wait_tensorcnt(1); // oldest DMA done
            else              __builtin_amdgcn_s_wait_tensorcnt(0); // drain
        }
#else
        {
            const long long dwBase = (chunk0 + it) * CHUNK;
            for (int i = tid; i < CHUNK; i += BLOCK) {
                long long g = dwBase + i;
                sIdx[it & 1][i] = (g < totalDw) ? faces[g] : 0;
            }
        }
#endif
        __syncthreads(); // publish buffer it&1 to all 8 waves

        const long long f = (chunk0 + it) * BLOCK + tid;
        if (f < F) {
            const int* sidx = &sIdx[it & 1][0];
            const int i0 = sidx[3 * tid + 0];
            const int i1 = sidx[3 * tid + 1];
            const int i2 = sidx[3 * tid + 2];

            // Gathers: 12MB of vertex tables -> L2-resident (192MB L2).
            F3 ac = ldv(cano, i0), bc = ldv(cano, i1), cc = ldv(cano, i2);
            F3 ad = ldv(mesh, i0), bd = ldv(mesh, i1), cd = ldv(mesh, i2);

            F3 fn = f3cross(f3sub(cc, bc), f3sub(ac, bc));
            float area = 0.5f * sqrtf(f3dot(fn, fn));

            F3 Xc, Yc, Zc, Xd, Yd, Zd;
            triFrame(ac, bc, cc, Xc, Yc, Zc);
            triFrame(ad, bd, cd, Xd, Yd, Zd);

            // cano frame orthonormal => inv = transpose; Rt = Rd * Rc^T
            float m00 = Xd.x * Xc.x + Yd.x * Yc.x + Zd.x * Zc.x;
            float m01 = Xd.x * Xc.y + Yd.x * Yc.y + Zd.x * Zc.y;
            float m02 = Xd.x * Xc.z + Yd.x * Yc.z + Zd.x * Zc.z;
            float m10 = Xd.y * Xc.x + Yd.y * Yc.x + Zd.y * Zc.x;
            float m11 = Xd.y * Xc.y + Yd.y * Yc.y + Zd.y * Zc.y;
            float m12 = Xd.y * Xc.z + Yd.y * Yc.z + Zd.y * Zc.z;
            float m20 = Xd.z * Xc.x + Yd.z * Yc.x + Zd.z * Zc.x;
            float m21 = Xd.z * Xc.y + Yd.z * Yc.y + Zd.z * Zc.y;
            float m22 = Xd.z * Xc.z + Yd.z * Yc.z + Zd.z * Zc.z;

            float t0 = 1.0f + m00 + m11 + m22;
            float t1 = 1.0f + m00 - m11 - m22;
            float t2 = 1.0f - m00 + m11 - m22;
            float t3 = 1.0f - m00 - m11 + m22;
            float q0 = t0 > 0.0f ? sqrtf(t0) : 0.0f;
            float q1 = t1 > 0.0f ? sqrtf(t1) : 0.0f;
            float q2 = t2 > 0.0f ? sqrtf(t2) : 0.0f;
            float q3 = t3 > 0.0f ? sqrtf(t3) : 0.0f;

            int idx = 0;
            float best = q0;
            if (q1 > best) { best = q1; idx = 1; }
            if (q2 > best) { best = q2; idx = 2; }
            if (q3 > best) { best = q3; idx = 3; }

            float d21 = m21 - m12, d02 = m02 - m20, d10 = m10 - m01;
            float s10 = m10 + m01, s02 = m02 + m20, s12 = m12 + m21;

            float qx, qy, qz, qw;
            if (idx == 0)      { qx = q0 * q0; qy = d21;     qz = d02;     qw = d10; }
            else if (idx == 1) { qx = d21;     qy = q1 * q1; qz = s10;     qw = s02; }
            else if (idx == 2) { qx = d02;     qy = s10;     qz = q2 * q2; qw = s12; }
            else               { qx = d10;     qy = s02;     qz = s12;     qw = q3 * q3; }

            float s = area / (2.0f * fmaxf(best, 0.1f));
            float w0 = qx * s, w1 = qy * s, w2 = qz * s, w3 = qw * s;

            float* p0 = out + 4ull * (unsigned)i0;
            float* p1 = out + 4ull * (unsigned)i1;
            float* p2 = out + 4ull * (unsigned)i2;
            atomicAdd(p0 + 0, w0); atomicAdd(p0 + 1, w1); atomicAdd(p0 + 2, w2); atomicAdd(p0 + 3, w3);
            atomicAdd(p1 + 0, w0); atomicAdd(p1 + 1, w1); atomicAdd(p1 + 2, w2); atomicAdd(p1 + 3, w3);
            atomicAdd(p2 + 0, w0); atomicAdd(p2 + 1, w1); atomicAdd(p2 + 2, w2); atomicAdd(p2 + 3, w3);
        }
        __syncthreads(); // all waves done reading buffer it&1

#if USE_TDM
        if (tid < 32 && it + 2 < nit) { // refill the freed buffer
            tdm_issue(faces, (chunk0 + it + 2) * CHUNK, totalDw,
                      ((it & 1) == 0) ? ldsOff0 : ldsOff1);
        }
#endif
    }
}

// ---- kernel 3: per-vertex normalize (in place, b128) ------------------------
__global__ void pvq_normalize_kernel(float* __restrict__ out, int V) {
    int v = blockIdx.x * blockDim.x + threadIdx.x;
    if (v >= V) return;
    float4* p = reinterpret_cast<float4*>(out);
    float4 q = p[v];
    float n = sqrtf(q.x * q.x + q.y * q.y + q.z * q.z + q.w * q.w);
    float r = 1.0f / fmaxf(n, 1e-6f);
    q.x *= r; q.y *= r; q.z *= r; q.w *= r;
    p[v] = q;
}

extern "C" void kernel_launch(void* const* d_in, const int* in_sizes, int n_in,
                              void* d_out, int out_size, void* d_ws, size_t ws_size,
                              hipStream_t stream) {
    const float* mesh = (const float*)d_in[0]; // mesh_verts (V,3) f32
    const float* cano = (const float*)d_in[1]; // cano_verts (V,3) f32
    const int* faces = (const int*)d_in[2];    // cano_faces (F,3) i32
    float* out = (float*)d_out;                // (V,4) f32

    const int V = in_sizes[0] / 3;
    const int F = in_sizes[2] / 3;

    const int zn = V * 4;
    pvq_zero_kernel<<<(zn + 255) / 256, 256, 0, stream>>>(out, zn);

    const long long totalChunks = ((long long)F + BLOCK - 1) / BLOCK;
    const int nBlocks = (int)((totalChunks + NIT - 1) / NIT);
    pvq_face_kernel<<<nBlocks, BLOCK, 0, stream>>>(mesh, cano, faces, out, F);

    pvq_normalize_kernel<<<(V + 255) / 256, 256, 0, stream>>>(out, V);
}